// BaseDomainBatchNorm_21861383536853
// MI455X (gfx1250) — compile-verified
//
#include <hip/hip_runtime.h>

#define C_CH   512
#define N_DOM  16
#define EPSV   1e-5f
#define NCHUNK 32   // row chunks for stats partials

typedef __attribute__((ext_vector_type(2))) float v2f;
typedef __attribute__((ext_vector_type(8))) float v8f;

// ---------------------------------------------------------------- counts ----
__global__ void k_init_cnt(int* cnt) {
    if (threadIdx.x < N_DOM) cnt[threadIdx.x] = 0;
}

__global__ void k_count(const int* __restrict__ d, int* __restrict__ cnt, int n) {
    __shared__ int h[N_DOM];
    if (threadIdx.x < N_DOM) h[threadIdx.x] = 0;
    __syncthreads();
    const int i = blockIdx.x * blockDim.x + threadIdx.x;
    if (i < n) atomicAdd(&h[d[i]], 1);        // ds_add_u32 (deterministic)
    __syncthreads();
    if (threadIdx.x < N_DOM) atomicAdd(&cnt[threadIdx.x], h[threadIdx.x]);
}

// ------------------------------------------------- stats via f32 WMMA -------
// S[16,512] = OneHot(d)^T * X ; SQ = OneHot^T * (X*X), done as 16x16x4 WMMAs.
// A (16x4): lanes 0-15 hold M=lane; vgpr0 = K0 (lanes<16) / K2 (lanes>=16),
//           vgpr1 = K1 / K3.  B (4x16) mirrors: vgpr0 = row K0/K2, vgpr1 = K1/K3.
__global__ void __launch_bounds__(256) k_stats(const float* __restrict__ X,
                                               const int* __restrict__ d,
                                               float* __restrict__ Sp,
                                               float* __restrict__ SQp,
                                               int n) {
    const int tid  = threadIdx.x;
    const int wave = tid >> 5;
    const int lane = tid & 31;
    const int half = lane >> 4;     // 0: K0/K1 rows, 1: K2/K3 rows
    const int m    = lane & 15;     // A: domain row; B: channel column
    const int cbase = blockIdx.x * 16;

    const int rows_per_block = n / gridDim.y;     // 2048 for N=65536
    const int rows_per_wave  = rows_per_block >> 3;
    const int r0 = blockIdx.y * rows_per_block + wave * rows_per_wave;

    v8f acc_s = {0.f,0.f,0.f,0.f,0.f,0.f,0.f,0.f};
    v8f acc_q = {0.f,0.f,0.f,0.f,0.f,0.f,0.f,0.f};

    #pragma unroll 4
    for (int base = r0; base < r0 + rows_per_wave; base += 4) {
        const int rA = base + (half << 1);
        const int rB = rA + 1;
        const int dA = d[rA];
        const int dB = d[rB];
        v2f a, b;
        a.x = (dA == m) ? 1.0f : 0.0f;     // one-hot A matrix
        a.y = (dB == m) ? 1.0f : 0.0f;
        b.x = X[(size_t)rA * C_CH + cbase + m];
        b.y = X[(size_t)rB * C_CH + cbase + m];
        v2f b2 = b * b;
        acc_s = __builtin_amdgcn_wmma_f32_16x16x4_f32(false, a, false, b,
                                                      (short)0, acc_s, false, false);
        acc_q = __builtin_amdgcn_wmma_f32_16x16x4_f32(false, a, false, b2,
                                                      (short)0, acc_q, false, false);
    }

    // deterministic cross-wave reduction (no float atomics)
    __shared__ float red[8][2][16][16];           // 16 KB
    #pragma unroll
    for (int r = 0; r < 8; ++r) {                 // C/D layout: vgpr r, dom = r + half*8
        red[wave][0][r + half * 8][m] = acc_s[r];
        red[wave][1][r + half * 8][m] = acc_q[r];
    }
    __syncthreads();
    const int dom = tid >> 4;
    const int ch  = tid & 15;
    float s = 0.f, q = 0.f;
    #pragma unroll
    for (int w = 0; w < 8; ++w) {
        s += red[w][0][dom][ch];
        q += red[w][1][dom][ch];
    }
    const int out = (blockIdx.y * N_DOM + dom) * C_CH + cbase + ch;
    Sp[out]  = s;
    SQp[out] = q;
}

// ---------------------------------------------------------------- finalize --
__global__ void k_finalize(const float* __restrict__ Sp, const float* __restrict__ SQp,
                           const int* __restrict__ cnt,
                           const float* __restrict__ gamma, const float* __restrict__ beta,
                           float* __restrict__ scale, float* __restrict__ shift) {
    const int o   = blockIdx.x * blockDim.x + threadIdx.x;   // 0 .. 16*512-1
    const int dom = o / C_CH;
    const int c   = o % C_CH;
    float s = 0.f, q = 0.f;
    #pragma unroll 8
    for (int ch = 0; ch < NCHUNK; ++ch) {
        s += Sp [(ch * N_DOM + dom) * C_CH + c];
        q += SQp[(ch * N_DOM + dom) * C_CH + c];
    }
    float nf   = fmaxf((float)cnt[dom], 1.0f);
    float mean = s / nf;
    float var  = q / nf - mean * mean;
    float inv  = rsqrtf(var + EPSV);
    float sc   = inv * gamma[o];
    scale[o] = sc;
    shift[o] = beta[o] - mean * sc;
}

// --------------------------------------------------------------- normalize --
__global__ void __launch_bounds__(256) k_norm(const float* __restrict__ X,
                                              const int* __restrict__ d,
                                              const float* __restrict__ scale,
                                              const float* __restrict__ shift,
                                              float* __restrict__ Y, int n) {
    __shared__ float4 s_sc[N_DOM * 32];   // 16 dom x 128 ch (as float4) = 8 KB
    __shared__ float4 s_sh[N_DOM * 32];
    const int tid = threadIdx.x;
    const int ct  = blockIdx.x;           // 128-channel tile
    const float4* scG = (const float4*)scale;
    const float4* shG = (const float4*)shift;
    #pragma unroll
    for (int j = 0; j < 2; ++j) {
        const int idx = j * 256 + tid;    // 0..511
        const int dm  = idx >> 5, c4 = idx & 31;
        s_sc[idx] = scG[dm * (C_CH / 4) + ct * 32 + c4];
        s_sh[idx] = shG[dm * (C_CH / 4) + ct * 32 + c4];
    }
    __syncthreads();

    const int rows_per_block = n / gridDim.y;   // 128
    const int r0 = blockIdx.y * rows_per_block;
    const int c4 = tid & 31;                    // wave = one row (32 x float4 = 128 ch)
    const int cw = C_CH / 4;
    const float4* X4 = (const float4*)X;
    float4* Y4 = (float4*)Y;

    for (int it = 0; it < rows_per_block; it += 8) {
        const int row = r0 + it + (tid >> 5);
        const int dm  = d[row];                     // broadcast load
        const size_t off = (size_t)row * cw + ct * 32 + c4;
        const float4 x  = X4[off];
        const float4 sc = s_sc[dm * 32 + c4];       // ds_load_b128, conflict-free
        const float4 sh = s_sh[dm * 32 + c4];
        float4 y;
        y.x = fmaf(x.x, sc.x, sh.x);
        y.y = fmaf(x.y, sc.y, sh.y);
        y.z = fmaf(x.z, sc.z, sh.z);
        y.w = fmaf(x.w, sc.w, sh.w);
        Y4[off] = y;
    }
}

// ------------------------------------------------------------------ launch --
extern "C" void kernel_launch(void* const* d_in, const int* in_sizes, int n_in,
                              void* d_out, int out_size, void* d_ws, size_t ws_size,
                              hipStream_t stream) {
    const float* X     = (const float*)d_in[0];
    const int*   d     = (const int*)  d_in[1];
    // d_in[2] parameter_t, d_in[3] fm_mean: unused by reference
    const float* gamma = (const float*)d_in[4];
    const float* beta  = (const float*)d_in[5];
    float* Y = (float*)d_out;
    const int n = in_sizes[1];                     // 65536 rows

    // workspace layout (needs ~2.1 MB)
    float* Sp    = (float*)d_ws;                               // [NCHUNK][16][512]
    float* SQp   = Sp  + (size_t)NCHUNK * N_DOM * C_CH;        // [NCHUNK][16][512]
    float* scale = SQp + (size_t)NCHUNK * N_DOM * C_CH;        // [16][512]
    float* shift = scale + N_DOM * C_CH;                       // [16][512]
    int*   cnt   = (int*)(shift + N_DOM * C_CH);               // [16]

    k_init_cnt<<<1, 32, 0, stream>>>(cnt);
    k_count<<<(n + 255) / 256, 256, 0, stream>>>(d, cnt, n);
    k_stats<<<dim3(C_CH / 16, NCHUNK), 256, 0, stream>>>(X, d, Sp, SQp, n);
    k_finalize<<<(N_DOM * C_CH) / 256, 256, 0, stream>>>(Sp, SQp, cnt, gamma, beta, scale, shift);
    k_norm<<<dim3(C_CH / 128, 512), 256, 0, stream>>>(X, d, scale, shift, Y, n);
}